// PostProcessor_26121991094523
// MI455X (gfx1250) — compile-verified
//
#include <hip/hip_runtime.h>
#include <math.h>
#include <stdint.h>

#define NREAL 1000
#define NP    1024          // padded proposals (power of two for sort / bit tricks)
#define NC    81
#define NCLS  80            // foreground classes
#define IMGW  1333.0f
#define IMGH  800.0f
#define SCORE_TH 0.05f
#define NMS_TH   0.5f
#define DETS  100
#define BBOX_CLIP 4.1351665567423563f   // log(1000/16)
#define PAD_SENTINEL (-1.0e30f)

typedef float v2f __attribute__((ext_vector_type(2)));
typedef float v8f __attribute__((ext_vector_type(8)));
typedef unsigned int u32x4 __attribute__((ext_vector_type(4)));
typedef int i32x4 __attribute__((ext_vector_type(4)));
typedef int i32x8 __attribute__((ext_vector_type(8)));

#if defined(__has_builtin)
#if __has_builtin(__builtin_amdgcn_wmma_f32_16x16x4_f32)
#define HAVE_WMMA4 1
#endif
#if __has_builtin(__builtin_amdgcn_tensor_load_to_lds) && \
    __has_builtin(__builtin_amdgcn_s_wait_tensorcnt)
#define HAVE_TDM 1
#endif
#endif

// ---------------------------------------------------------------------------
// K1: per-proposal softmax (81 classes) + box decode + clip.
// Writes class-major scores[80][1024] and boxes[80][1024][4] (original order).
// ---------------------------------------------------------------------------
__global__ __launch_bounds__(256) void k1_decode_softmax(
    const float* __restrict__ logits, const float* __restrict__ breg,
    const float* __restrict__ props, float* __restrict__ scores,
    float* __restrict__ boxes)
{
    int i = blockIdx.x * 256 + threadIdx.x;
    if (i >= NP) return;
    if (i < NREAL) {
        const float* lg = logits + (size_t)i * NC;
        const float* r  = breg + (size_t)i * (4 * NC);
        __builtin_prefetch(r, 0, 1);      // global_prefetch_b8
        float mx = lg[0];
        for (int c = 1; c < NC; ++c) mx = fmaxf(mx, lg[c]);
        float sum = 0.f;
        for (int c = 0; c < NC; ++c) sum += expf(lg[c] - mx);
        float inv = 1.f / sum;

        float4 p = ((const float4*)props)[i];
        float w  = p.z - p.x + 1.f, h = p.w - p.y + 1.f;
        float cx = p.x + 0.5f * w,  cy = p.y + 0.5f * h;

        for (int c = 1; c < NC; ++c) {
            int cc = c - 1;
            float dx = r[c*4+0] / 10.0f;
            float dy = r[c*4+1] / 10.0f;
            float dw = fminf(r[c*4+2] / 5.0f, BBOX_CLIP);
            float dh = fminf(r[c*4+3] / 5.0f, BBOX_CLIP);
            float pcx = dx*w + cx, pcy = dy*h + cy;
            float pw  = expf(dw) * w, ph = expf(dh) * h;
            float x1 = fminf(fmaxf(pcx - 0.5f*pw,        0.f), IMGW - 1.f);
            float y1 = fminf(fmaxf(pcy - 0.5f*ph,        0.f), IMGH - 1.f);
            float x2 = fminf(fmaxf(pcx + 0.5f*pw - 1.f,  0.f), IMGW - 1.f);
            float y2 = fminf(fmaxf(pcy + 0.5f*ph - 1.f,  0.f), IMGH - 1.f);
            size_t bo = ((size_t)cc * NP + i) * 4;
            boxes[bo+0]=x1; boxes[bo+1]=y1; boxes[bo+2]=x2; boxes[bo+3]=y2;
            scores[(size_t)cc * NP + i] = expf(lg[c] - mx) * inv;
        }
    } else {
        for (int cc = 0; cc < NCLS; ++cc) {
            scores[(size_t)cc * NP + i] = PAD_SENTINEL;
            size_t bo = ((size_t)cc * NP + i) * 4;
            boxes[bo+0]=0.f; boxes[bo+1]=0.f; boxes[bo+2]=0.f; boxes[bo+3]=0.f;
        }
    }
}

// ---------------------------------------------------------------------------
// K2: per-class bitonic sort of 1024 (score, idx), descending score,
// ties -> lower index (matches stable jnp.argsort(-scores)).
// Also gathers sorted boxes for the NMS kernel.
// ---------------------------------------------------------------------------
__global__ __launch_bounds__(1024) void k2_sort(
    const float* __restrict__ scores, const float* __restrict__ boxes,
    float* __restrict__ sScore, int* __restrict__ order,
    float* __restrict__ sBoxes)
{
    __shared__ float ss[NP];
    __shared__ int   si[NP];
    int c = blockIdx.x, tid = threadIdx.x;
    ss[tid] = scores[(size_t)c * NP + tid];
    si[tid] = tid;
    __syncthreads();
    for (int k = 2; k <= NP; k <<= 1) {
        for (int j = k >> 1; j > 0; j >>= 1) {
            int ixj = tid ^ j;
            if (ixj > tid) {
                float s0 = ss[tid], s1 = ss[ixj];
                int   i0 = si[tid], i1 = si[ixj];
                bool before01 = (s0 > s1) || (s0 == s1 && i0 < i1);
                bool ascending = ((tid & k) == 0);
                bool sw = ascending ? !before01 : before01;
                if (sw) { ss[tid]=s1; ss[ixj]=s0; si[tid]=i1; si[ixj]=i0; }
            }
            __syncthreads();
        }
    }
    sScore[(size_t)c * NP + tid] = ss[tid];
    order [(size_t)c * NP + tid] = si[tid];
    float4 b4 = ((const float4*)boxes)[(size_t)c * NP + si[tid]];
    ((float4*)sBoxes)[(size_t)c * NP + tid] = b4;
}

// ---------------------------------------------------------------------------
// K3: per-class NMS. One workgroup (32 wave32s) per class; 152 KB LDS
// (2 workgroups fit in CDNA5's 320 KB WGP LDS).
//  - Sorted boxes (16 KB) DMA'd global->LDS by the Tensor Data Mover
//    (tensor_load_to_lds + s_wait_tensorcnt), cooperative-copy fallback.
//  - Each wave computes a 16x16 IoU tile: area_i+area_j via
//    V_WMMA_F32_16X16X4_F32 (co-executes with VALU min/max intersection);
//    IoU>0.5 test done division-free; a wave32 ballot packs two mask rows,
//    stored with one predicated ds_store_b16.
//  - Wave 0 runs the greedy serial scan from LDS, each lane owning one
//    32-bit suppression word in a VGPR.
// ---------------------------------------------------------------------------
__global__ __launch_bounds__(1024) void k3_nms(
    const float* __restrict__ sScore, const int* __restrict__ order,
    const float* __restrict__ sBoxes, float* __restrict__ masked)
{
    __shared__ unsigned int maskS[NP][32];   // 128 KB suppression bitmatrix
    __shared__ float4 bxS[NP];               //  16 KB sorted boxes
    __shared__ float scL[NP];                //   4 KB sorted scores
    __shared__ int   orL[NP];                //   4 KB sort permutation

    int c = blockIdx.x, tid = threadIdx.x;
    int wave = tid >> 5, lane = tid & 31, lr = lane & 15;
    const float4* RB = (const float4*)sBoxes + (size_t)c * NP;

    scL[tid] = sScore[(size_t)c * NP + tid];
    orL[tid] = order [(size_t)c * NP + tid];
    #pragma unroll
    for (int w = 0; w < 32; ++w) maskS[tid][w] = 0u;

#ifdef HAVE_TDM
    if (wave == 0) {
        // D# descriptor (CDNA5 ISA): 1-D tile, 16 KB,
        // data_size=8B (2048 elements), groups 2/3 zero (<=2D tensor).
        unsigned long long ga = (unsigned long long)(uintptr_t)RB;
        unsigned ldsa = (unsigned)(uintptr_t)&bxS[0];
        u32x4 g0;
        g0[0] = 1u;                                            // count=1, user mode
        g0[1] = ldsa;                                          // lds_addr
        g0[2] = (unsigned)(ga & 0xFFFFFFFFull);                // global_addr lo
        g0[3] = (unsigned)((ga >> 32) & 0x01FFFFFFull)         // global_addr hi
              | 0x80000000u;                                   // type=2 ("image")
        i32x8 g1;
        g1[0] = (int)(3u << 16);        // data_size=8B; workgroup_mask=0
        g1[1] = (int)(2048u << 16);     // tensor_dim0 = 2048 (bits 79:48)
        g1[2] = (int)(1u << 16);        // tensor_dim1 = 1    (bits 111:80)
        g1[3] = (int)(2048u << 16);     // tile_dim0 = 2048   (bits 127:112)
        g1[4] = 1;                      // tile_dim1 = 1      (bits 143:128)
        g1[5] = 2048;                   // tensor_dim0_stride lo32
        g1[6] = (int)(2048u << 16);     // tensor_dim1_stride lo16 (bits 223:208)
        g1[7] = 0;
        i32x4 z4 = {0, 0, 0, 0};
#if __clang_major__ >= 23
        i32x8 z8 = {0, 0, 0, 0, 0, 0, 0, 0};
        __builtin_amdgcn_tensor_load_to_lds(g0, g1, z4, z4, z8, 0);
#else
        __builtin_amdgcn_tensor_load_to_lds(g0, g1, z4, z4, 0);
#endif
        __builtin_amdgcn_s_wait_tensorcnt(0);
    }
#else
    bxS[tid] = RB[tid];
#endif
    __syncthreads();

    bool lo = lane < 16;
    const int NTILES = (64 * 65) / 2;       // cT >= rT only (j > i impossible elsewhere)
    for (int t = wave; t < NTILES; t += 32) {
        int rT = 0, tt = t;
        while (tt >= 64 - rT) { tt -= 64 - rT; ++rT; }
        int cT = rT + tt;

        float4 cb = bxS[cT * 16 + lr];          // this lane's column box
        float4 ra = bxS[rT * 16 + lr];          // row box for A-matrix build
        float areaC = (cb.z - cb.x + 1.f) * (cb.w - cb.y + 1.f);
        float areaR = (ra.z - ra.x + 1.f) * (ra.w - ra.y + 1.f);

        v8f aSum;
#ifdef HAVE_WMMA4
        // A(16x4): lanes 0-15 carry K0=area_row, K1=1; lanes 16-31 (K2,K3)=0
        // B(4x16): lanes 0-15 carry K0=1, K1=area_col; lanes 16-31 (K2,K3)=0
        // => D[m,n] = area_row[m] + area_col[n]
        v2f av, bvv;
        av.x  = lo ? areaR : 0.f;  av.y  = lo ? 1.f   : 0.f;
        bvv.x = lo ? 1.f   : 0.f;  bvv.y = lo ? areaC : 0.f;
        v8f cz = {};
        aSum = __builtin_amdgcn_wmma_f32_16x16x4_f32(
            false, av, false, bvv, (short)0, cz, false, false);
#endif
        int hi8 = (lane >> 4) << 3;    // D-layout: lane holds rows v+hi8, col lr
        #pragma unroll
        for (int v = 0; v < 8; ++v) {
            int ml = v + hi8;
            float4 rb = bxS[rT * 16 + ml];      // row box (broadcast ds_load_b128)
            float iw = fminf(rb.z, cb.z) - fmaxf(rb.x, cb.x) + 1.f;
            float ih = fminf(rb.w, cb.w) - fmaxf(rb.y, cb.y) + 1.f;
            iw = fmaxf(iw, 0.f); ih = fmaxf(ih, 0.f);
            float inter = iw * ih;
#ifdef HAVE_WMMA4
            float asv = aSum[v];
#else
            float asv = (rb.z - rb.x + 1.f) * (rb.w - rb.y + 1.f) + areaC;
#endif
            // iou > 0.5  <=>  inter > 0.5*(areaSum - inter)   (denominator >= 1)
            int row = rT * 16 + ml, col = cT * 16 + lr;
            bool bit = (inter > 0.5f * (asv - inter)) && (col > row);
            unsigned bal = (unsigned)__ballot(bit);
            // lane 0 stores rows rT*16+v (bal[15:0]); lane 16 rows rT*16+v+8 (bal[31:16])
            if ((lane & 15) == 0) {
                unsigned short hw =
                    (unsigned short)(lo ? (bal & 0xFFFFu) : (bal >> 16));
                ((unsigned short*)&maskS[0][0])[row * 64 + cT] = hw;
            }
        }
    }
    __syncthreads();

    // serial greedy scan by wave 0; lane w owns suppression word w in a VGPR
    if (tid < 32) {
        unsigned supW = 0u;
        for (int i = 0; i < NP; ++i) {
            unsigned sw = (unsigned)__shfl((int)supW, i >> 5);
            bool supb = (sw >> (i & 31)) & 1u;
            float sc = scL[i];
            bool keep = (sc > SCORE_TH) && !supb;
            if (keep) supW |= maskS[i][tid];
            if (tid == 0) {
                int oi = orL[i];
                masked[(size_t)c * NP + oi] =
                    (oi >= NREAL) ? PAD_SENTINEL : (keep ? sc : -1.0f);
            }
        }
    }
}

// ---------------------------------------------------------------------------
// K4: top-100 over masked[80][1024] (values descending, ties -> lowest flat
// index, matching jax.lax.top_k). Iterative extraction, single workgroup.
// ---------------------------------------------------------------------------
__global__ __launch_bounds__(1024) void k4_topk(
    float* __restrict__ masked, const float* __restrict__ boxes,
    float* __restrict__ out)
{
    __shared__ float vb[NP];
    __shared__ int   ib[NP];
    int tid = threadIdx.x;
    for (int k = 0; k < DETS; ++k) {
        float bv = -3.0e38f; int bi = 0;
        for (int t = tid; t < NCLS * NP; t += NP) {
            float v = masked[t];
            if (v > bv) { bv = v; bi = t; }   // strict > keeps smallest index on ties
        }
        vb[tid] = bv; ib[tid] = bi;
        __syncthreads();
        for (int s = NP >> 1; s > 0; s >>= 1) {
            if (tid < s) {
                float v2 = vb[tid + s]; int i2 = ib[tid + s];
                if (v2 > vb[tid] || (v2 == vb[tid] && i2 < ib[tid])) {
                    vb[tid] = v2; ib[tid] = i2;
                }
            }
            __syncthreads();
        }
        if (tid == 0) {
            int wi = ib[0]; float wv = vb[0];
            int cc = wi >> 10, i = wi & (NP - 1);
            const float* bp = boxes + ((size_t)cc * NP + i) * 4;
            out[k*6+0] = bp[0]; out[k*6+1] = bp[1];
            out[k*6+2] = bp[2]; out[k*6+3] = bp[3];
            out[k*6+4] = wv;    out[k*6+5] = (float)(cc + 1);
            masked[wi] = -3.0e38f;          // consume (ws is rebuilt by K3 each call)
        }
        __syncthreads();
    }
}

// ---------------------------------------------------------------------------
extern "C" void kernel_launch(void* const* d_in, const int* in_sizes, int n_in,
                              void* d_out, int out_size, void* d_ws, size_t ws_size,
                              hipStream_t stream) {
    const float* logits = (const float*)d_in[0];   // [1000, 81]
    const float* breg   = (const float*)d_in[1];   // [1000, 324]
    const float* props  = (const float*)d_in[2];   // [1000, 4]

    float* ws     = (float*)d_ws;                  // ~3.94 MB total
    float* scores = ws;                            // [80][1024]
    float* boxes  = scores + (size_t)NCLS * NP;    // [80][1024][4]
    float* sScore = boxes  + (size_t)NCLS * NP * 4;
    int*   order  = (int*)(sScore + (size_t)NCLS * NP);
    float* sBoxes = (float*)(order + (size_t)NCLS * NP);
    float* masked = sBoxes + (size_t)NCLS * NP * 4;

    k1_decode_softmax<<<NP / 256, 256, 0, stream>>>(logits, breg, props, scores, boxes);
    k2_sort<<<NCLS, NP, 0, stream>>>(scores, boxes, sScore, order, sBoxes);
    k3_nms <<<NCLS, NP, 0, stream>>>(sScore, order, sBoxes, masked);
    k4_topk<<<1, NP, 0, stream>>>(masked, boxes, (float*)d_out);

    (void)in_sizes; (void)n_in; (void)out_size; (void)ws_size;
}